// Trigger_Recognition_23407571763854
// MI455X (gfx1250) — compile-verified
//
#include <hip/hip_runtime.h>

typedef float v2f __attribute__((ext_vector_type(2)));
typedef float v8f __attribute__((ext_vector_type(8)));

constexpr int T_LEN  = 16384;
constexpr int HID    = 256;
constexpr int EMB    = 16;
constexpr int TMPD   = 128;
constexpr int EVD    = 2048;
constexpr int W1LD   = HID + EMB;   // 272
constexpr float NEG_SLOPE = 0.01f;

// ---------------------------------------------------------------------------
// WMMA f32 16x16x4 fragment loaders (per ISA 7.12.2 layouts, wave32).
// A (16x4, MxK): lanes 0-15 -> M=lane, K={kk,kk+1}; lanes 16-31 -> K={kk+2,kk+3}
// B (4x16, KxN): lane n=lane&15 holds column n, same K split as A.
// Here B is sourced transposed from a row-major weight W[row=n][k].
// ---------------------------------------------------------------------------
__device__ __forceinline__ v2f load_a_frag(const float* __restrict__ A, int lda,
                                           int row0, int kk, int lane) {
    int m = row0 + (lane & 15);
    int k = kk + ((lane & 16) >> 3);           // +2 for upper half-wave
    const float* p = A + (size_t)m * lda + k;
    v2f r; r.x = p[0]; r.y = p[1]; return r;
}

__device__ __forceinline__ v2f load_bT_frag(const float* __restrict__ W, int ldw,
                                            int col0, int kbase, int kk, int lane) {
    int n = col0 + (lane & 15);
    int k = kbase + kk + ((lane & 16) >> 3);
    const float* p = W + (size_t)n * ldw + k;
    v2f r; r.x = p[0]; r.y = p[1]; return r;
}

// ---------------------------------------------------------------------------
// Phase 1: pre_h[t][j] = sum_k X[t][k]*W1[j][k] + b1[j]   (16384x256x128 GEMM)
// One 16x16 output tile per wave, 8 waves / 256-thread block.
// ---------------------------------------------------------------------------
__global__ __launch_bounds__(256) void gemm_pre_kernel(
    const float* __restrict__ X, const float* __restrict__ W1,
    const float* __restrict__ b1, float* __restrict__ pre_h) {
    const int lane = threadIdx.x & 31;
    const int wave = threadIdx.x >> 5;
    const int tile = blockIdx.x * 8 + wave;
    const int tn   = TMPD / 16;                // 8 column tiles
    const int row0 = (tile / tn) * 16;
    const int col0 = (tile % tn) * 16;

    v8f c = {};
#pragma unroll 8
    for (int kk = 0; kk < HID; kk += 4) {
        v2f a = load_a_frag(X, HID, row0, kk, lane);
        v2f b = load_bT_frag(W1, W1LD, col0, 0, kk, lane);
        c = __builtin_amdgcn_wmma_f32_16x16x4_f32(
                /*neg_a=*/false, a, /*neg_b=*/false, b,
                /*c_mod=*/(short)0, c, /*reuse_a=*/false, /*reuse_b=*/false);
    }

    const int moff = (lane & 16) >> 1;         // +8 rows for upper half-wave
    const int col  = col0 + (lane & 15);
    const float bias = b1[col];
#pragma unroll
    for (int r = 0; r < 8; ++r)
        pre_h[(size_t)(row0 + r + moff) * TMPD + col] = c[r] + bias;
}

// ---------------------------------------------------------------------------
// Phase 2: E2H[e][j] = sum_m emb[e][m]*W1[j][256+m]   (2048x16x128 GEMM)
// Precomputes the event-embedding contribution to h for every vocab entry,
// turning the serial loop's embedding matmul into a row gather.
// ---------------------------------------------------------------------------
__global__ __launch_bounds__(256) void gemm_e2h_kernel(
    const float* __restrict__ emb, const float* __restrict__ W1,
    float* __restrict__ e2h) {
    const int lane = threadIdx.x & 31;
    const int wave = threadIdx.x >> 5;
    const int tile = blockIdx.x * 8 + wave;
    const int tn   = TMPD / 16;
    const int row0 = (tile / tn) * 16;
    const int col0 = (tile % tn) * 16;

    v8f c = {};
#pragma unroll
    for (int kk = 0; kk < EMB; kk += 4) {
        v2f a = load_a_frag(emb, EMB, row0, kk, lane);
        v2f b = load_bT_frag(W1, W1LD, col0, HID, kk, lane);
        c = __builtin_amdgcn_wmma_f32_16x16x4_f32(
                false, a, false, b, (short)0, c, false, false);
    }

    const int moff = (lane & 16) >> 1;
    const int col  = col0 + (lane & 15);
#pragma unroll
    for (int r = 0; r < 8; ++r)
        e2h[(size_t)(row0 + r + moff) * TMPD + col] = c[r];
}

// ---------------------------------------------------------------------------
// Phase 3: serial autoregressive decode. Single persistent workgroup
// (1024 threads = 32 wave32 waves on one WGP). W2 (1MB) stays L2-resident.
// Per step: h(128) -> LDS, 2 logits/thread (b128 loads), wave32 shfl_xor
// reductions for argmax + exp-sum, log-softmax writeout, argmax feedback.
// ---------------------------------------------------------------------------
__global__ __launch_bounds__(1024) void decode_kernel(
    const float* __restrict__ pre_h, const float* __restrict__ e2h,
    const float* __restrict__ W2, const float* __restrict__ b2,
    const float* __restrict__ emb, const int* __restrict__ init_idx,
    float* __restrict__ out_lp, float* __restrict__ out_emb,
    float* __restrict__ out_idx) {
    __shared__ float h_sh[TMPD];
    __shared__ float red_val[32];
    __shared__ int   red_idx[32];
    __shared__ float red_sum[32];

    const int tid  = threadIdx.x;
    const int lane = tid & 31;
    const int wave = tid >> 5;
    const int row0 = tid;              // logit rows handled by this thread
    const int row1 = tid + 1024;

    const float4* __restrict__ w0 = (const float4*)(W2 + (size_t)row0 * TMPD);
    const float4* __restrict__ w1 = (const float4*)(W2 + (size_t)row1 * TMPD);
    const float  bias0 = b2[row0];
    const float  bias1 = b2[row1];

    int idx = init_idx[0];

    for (int t = 0; t < T_LEN; ++t) {
        // h = leaky_relu(pre_h[t] + E2H[idx])  (b1 already folded into pre_h)
        if (tid < TMPD) {
            float x = pre_h[(size_t)t * TMPD + tid] + e2h[(size_t)idx * TMPD + tid];
            h_sh[tid] = (x > 0.0f) ? x : NEG_SLOPE * x;
        }
        if (tid == 0 && t + 1 < T_LEN)
            __builtin_prefetch(pre_h + (size_t)(t + 1) * TMPD, 0, 1);
        __syncthreads();

        // two logits per thread: dot(W2[row], h) with b128 loads
        const float4* hs = (const float4*)h_sh;
        float l0 = bias0, l1 = bias1;
#pragma unroll 8
        for (int j = 0; j < TMPD / 4; ++j) {
            float4 hv = hs[j];
            float4 a  = w0[j];
            float4 b  = w1[j];
            l0 += a.x * hv.x + a.y * hv.y + a.z * hv.z + a.w * hv.w;
            l1 += b.x * hv.x + b.y * hv.y + b.z * hv.z + b.w * hv.w;
        }

        // ---- block argmax (ties -> lowest index, matching jnp.argmax) ----
        float mval; int midx;
        if (l1 > l0) { mval = l1; midx = row1; } else { mval = l0; midx = row0; }
#pragma unroll
        for (int off = 16; off > 0; off >>= 1) {
            float ov = __shfl_xor(mval, off, 32);
            int   oi = __shfl_xor(midx, off, 32);
            if (ov > mval || (ov == mval && oi < midx)) { mval = ov; midx = oi; }
        }
        if (lane == 0) { red_val[wave] = mval; red_idx[wave] = midx; }
        __syncthreads();
        if (wave == 0) {
            mval = red_val[lane]; midx = red_idx[lane];
#pragma unroll
            for (int off = 16; off > 0; off >>= 1) {
                float ov = __shfl_xor(mval, off, 32);
                int   oi = __shfl_xor(midx, off, 32);
                if (ov > mval || (ov == mval && oi < midx)) { mval = ov; midx = oi; }
            }
            if (lane == 0) { red_val[0] = mval; red_idx[0] = midx; }
        }
        __syncthreads();
        const float gmax = red_val[0];
        const int   gidx = red_idx[0];

        // ---- block exp-sum for log-softmax ----
        float s = __expf(l0 - gmax) + __expf(l1 - gmax);
#pragma unroll
        for (int off = 16; off > 0; off >>= 1) s += __shfl_xor(s, off, 32);
        if (lane == 0) red_sum[wave] = s;
        __syncthreads();
        if (wave == 0) {
            s = red_sum[lane];
#pragma unroll
            for (int off = 16; off > 0; off >>= 1) s += __shfl_xor(s, off, 32);
            if (lane == 0) red_sum[0] = s;
        }
        __syncthreads();
        const float lse = gmax + __logf(red_sum[0]);

        // ---- outputs ----
        out_lp[(size_t)t * EVD + row0] = l0 - lse;
        out_lp[(size_t)t * EVD + row1] = l1 - lse;
        if (tid < EMB) out_emb[(size_t)t * EMB + tid] = emb[(size_t)gidx * EMB + tid];
        if (tid == 0)  out_idx[t] = (float)gidx;

        idx = gidx;          // greedy feedback
        __syncthreads();     // protect h_sh / reduction slots for next step
    }
}

// ---------------------------------------------------------------------------
extern "C" void kernel_launch(void* const* d_in, const int* in_sizes, int n_in,
                              void* d_out, int out_size, void* d_ws, size_t ws_size,
                              hipStream_t stream) {
    (void)in_sizes; (void)n_in; (void)out_size; (void)ws_size;
    const float* X        = (const float*)d_in[0];   // [T, 256]
    const float* emb      = (const float*)d_in[1];   // [2048, 16]
    const float* W1       = (const float*)d_in[2];   // [128, 272]
    const float* b1       = (const float*)d_in[3];   // [128]
    const float* W2       = (const float*)d_in[4];   // [2048, 128]
    const float* b2       = (const float*)d_in[5];   // [2048]
    const int*   init_idx = (const int*)d_in[6];     // [1]

    float* pre_h = (float*)d_ws;                       // T*128 f32 = 8 MB
    float* e2h   = pre_h + (size_t)T_LEN * TMPD;       // 2048*128 f32 = 1 MB

    float* out     = (float*)d_out;
    float* out_lp  = out;                              // [T, 2048]
    float* out_emb = out + (size_t)T_LEN * EVD;        // [T, 1, 16]
    float* out_idx = out_emb + (size_t)T_LEN * EMB;    // [T] (as float)

    // Phase 1: 8192 tiles of 16x16, 8 waves/block -> 1024 blocks
    gemm_pre_kernel<<<(T_LEN / 16) * (TMPD / 16) / 8, 256, 0, stream>>>(X, W1, b1, pre_h);
    // Phase 2: 1024 tiles -> 128 blocks
    gemm_e2h_kernel<<<(EVD / 16) * (TMPD / 16) / 8, 256, 0, stream>>>(emb, W1, e2h);
    // Phase 3: single persistent workgroup (serial chain)
    decode_kernel<<<1, 1024, 0, stream>>>(pre_h, e2h, W2, b2, emb, init_idx,
                                          out_lp, out_emb, out_idx);
}